// InvoiceGCN_8744553414859
// MI455X (gfx1250) — compile-verified
//
#include <hip/hip_runtime.h>
#include <hip/hip_bf16.h>

// ---------------------------------------------------------------------------
// ChebConv (K=3) x 4 layers for MI455X (gfx1250, wave32).
//  - graph scatter: L2-resident float4 gathers + native f32 atomics
//  - dense 3-term Chebyshev GEMM: V_WMMA_F32_16X16X4_F32, packed-B fragments,
//    MR M-tiles per wave (MR sized so accumulators never spill).
// ---------------------------------------------------------------------------

typedef __attribute__((ext_vector_type(2))) float v2f;
typedef __attribute__((ext_vector_type(8))) float v8f;

__device__ __forceinline__ void atomAddF(float* p, float v) {
    // lowers to global_atomic_add_f32 (no CAS loop)
    unsafeAtomicAdd(p, v);
}

// --------------------------- small utility kernels -------------------------

__global__ void k_zero4(float4* __restrict__ p, long long n4) {
    long long i = (long long)blockIdx.x * blockDim.x + threadIdx.x;
    if (i < n4) p[i] = (float4){0.f, 0.f, 0.f, 0.f};
}

__global__ void k_negcopy4(float4* __restrict__ d, const float4* __restrict__ s, long long n4) {
    long long i = (long long)blockIdx.x * blockDim.x + threadIdx.x;
    if (i < n4) {
        float4 v = s[i];
        d[i] = (float4){-v.x, -v.y, -v.z, -v.w};
    }
}

// edge_index is int64 but node ids fit in 32 bits: read low word only.
__global__ void k_deg(const int* __restrict__ src32, const float* __restrict__ w,
                      float* __restrict__ deg, int E) {
    int e = blockIdx.x * blockDim.x + threadIdx.x;
    if (e < E) atomAddF(&deg[src32[(long long)e << 1]], w[e]);
}

__global__ void k_dinv(float* __restrict__ deg, int n) {
    int i = blockIdx.x * blockDim.x + threadIdx.x;
    if (i < n) {
        float d = deg[i];
        deg[i] = (d > 0.0f) ? rsqrtf(d) : 0.0f;
    }
}

__global__ void k_edge_norm(const int* __restrict__ src32, const int* __restrict__ dst32,
                            const float* __restrict__ w, const float* __restrict__ dinv,
                            float* __restrict__ norm, int E) {
    int e = blockIdx.x * blockDim.x + threadIdx.x;
    if (e < E)
        norm[e] = dinv[src32[(long long)e << 1]] * w[e] * dinv[dst32[(long long)e << 1]];
}

// ------------------------------ sparse scatter -----------------------------
// out[dst[e], :] += scale * norm[e] * h[src[e], :]
// One thread per (edge, float4 chunk): coalesced gather, 4 fp32 atomics.
template <int F>
__global__ void k_scatter(const int* __restrict__ src32, const int* __restrict__ dst32,
                          const float* __restrict__ norm, const float* __restrict__ h,
                          float* __restrict__ out, float scale, int E) {
    constexpr int V = F / 4;  // float4 chunks per row (power of two)
    long long idx = (long long)blockIdx.x * blockDim.x + threadIdx.x;
    if (idx >= (long long)E * V) return;
    int e  = (int)(idx / V);
    int f4 = (int)(idx % V);
    float s = scale * norm[e];
    int   sn = src32[(long long)e << 1];
    int   dn = dst32[(long long)e << 1];
    const float4 hv = ((const float4*)(h + (long long)sn * F))[f4];
    float* o = out + (long long)dn * F + (long long)f4 * 4;
    atomAddF(o + 0, s * hv.x);
    atomAddF(o + 1, s * hv.y);
    atomAddF(o + 2, s * hv.z);
    atomAddF(o + 3, s * hv.w);
}

// ------------------------- weight fragment packing -------------------------
// Pack W[3, FIN, FOUT] into per-lane WMMA B fragments:
//   P2[ ((t*K4 + k4)*NT + nt)*32 + lane ] = { W[t][ka][col], W[t][ka+1][col] }
// with ka = k4*4 + (lane>>4)*2, col = nt*16 + (lane&15)  (0 beyond FOUT).
template <int FIN, int FOUT>
__global__ void k_pack_w(const float* __restrict__ W, float2* __restrict__ P2) {
    constexpr int NT = (FOUT + 15) / 16;
    constexpr int K4 = FIN / 4;
    int idx = blockIdx.x * blockDim.x + threadIdx.x;
    if (idx >= 3 * K4 * NT * 32) return;
    int lane = idx & 31;
    int rest = idx >> 5;
    int nt = rest % NT;  rest /= NT;
    int k4 = rest % K4;
    int t  = rest / K4;
    int ka  = k4 * 4 + (lane >> 4) * 2;
    int col = nt * 16 + (lane & 15);
    float x0 = 0.f, x1 = 0.f;
    if (col < FOUT) {
        const float* Wt = W + (long long)t * FIN * FOUT;
        x0 = Wt[(long long)ka * FOUT + col];
        x1 = Wt[(long long)(ka + 1) * FOUT + col];
    }
    P2[idx] = (float2){x0, x1};
}

// --------------------------- Chebyshev WMMA GEMM ---------------------------
// out[n, FOUT] = T0@W[0] + T1@W[1] + T2@W[2] + bias, optional ReLU.
// One wave handles MR consecutive 16-row tiles so each packed-B fragment
// feeds MR WMMAs.  MR chosen so accumulators fit in <=64 VGPRs (no spills).
// Tail tiles: loads clamped, stores guarded -> EXEC all-1s at every WMMA.
template <int FIN, int FOUT, bool RELU, int MR>
__global__ __launch_bounds__(128, 1) void k_cheb_gemm(
        const float* __restrict__ T0, const float* __restrict__ T1,
        const float* __restrict__ T2, const float* __restrict__ P,
        const float* __restrict__ bias, float* __restrict__ out, int n) {
    constexpr int NT = (FOUT + 15) / 16;
    constexpr int K4 = FIN / 4;
    const int lane   = threadIdx.x & 31;
    const int wave   = threadIdx.x >> 5;
    const int mtiles = (n + 15) >> 4;
    const int mt0    = (blockIdx.x * (blockDim.x >> 5) + wave) * MR;
    if (mt0 >= mtiles) return;  // wave-uniform

    const int colL  = lane & 15;
    const int khalf = (lane >> 4) * 2;  // lanes 0-15 -> K={0,1}, 16-31 -> K={2,3}

    int rows[MR];
#pragma unroll
    for (int m = 0; m < MR; ++m) {
        int mt = mt0 + m;
        if (mt > mtiles - 1) mt = mtiles - 1;  // clamp tail (store is guarded)
        rows[m] = mt * 16 + colL;
    }

    v8f acc[MR][NT];
#pragma unroll
    for (int m = 0; m < MR; ++m)
#pragma unroll
        for (int t = 0; t < NT; ++t)
            acc[m][t] = (v8f){0.f, 0.f, 0.f, 0.f, 0.f, 0.f, 0.f, 0.f};

    const float2* __restrict__ Pv = (const float2*)P;
    const float* Ts[3] = {T0, T1, T2};
#pragma unroll
    for (int t = 0; t < 3; ++t) {
        const float* __restrict__ T = Ts[t];
#pragma unroll 1  // keep the K-loop rolled: compact body, short live ranges
        for (int k4 = 0; k4 < K4; ++k4) {
            const int ka = k4 * 4 + khalf;
            v2f a[MR];
#pragma unroll
            for (int m = 0; m < MR; ++m) {
                const float2 av = *(const float2*)(T + (long long)rows[m] * FIN + ka);
                a[m].x = av.x;
                a[m].y = av.y;
            }
#pragma unroll
            for (int nt = 0; nt < NT; ++nt) {
                const float2 bv = Pv[(((t * K4 + k4) * NT + nt) << 5) + lane];
                v2f b;
                b.x = bv.x;
                b.y = bv.y;
#pragma unroll
                for (int m = 0; m < MR; ++m)
                    acc[m][nt] = __builtin_amdgcn_wmma_f32_16x16x4_f32(
                        /*neg_a=*/false, a[m], /*neg_b=*/false, b,
                        /*c_mod=*/(short)0, acc[m][nt],
                        /*reuse_a=*/false, /*reuse_b=*/false);
            }
        }
    }

    // Epilogue: C/D layout -> VGPR i: lanes 0-15 M=i, lanes 16-31 M=i+8
#pragma unroll
    for (int m = 0; m < MR; ++m) {
        const int mt = mt0 + m;
        if (mt < mtiles) {
#pragma unroll
            for (int nt = 0; nt < NT; ++nt) {
                const int col = nt * 16 + colL;
                if (col < FOUT) {
                    const float bv = bias[col];
#pragma unroll
                    for (int i = 0; i < 8; ++i) {
                        const int r = mt * 16 + i + (lane >> 4) * 8;
                        float v = acc[m][nt][i] + bv;
                        if (RELU) v = fmaxf(v, 0.0f);
                        out[(long long)r * FOUT + col] = v;
                    }
                }
            }
        }
    }
}

// ------------------------------- log_softmax -------------------------------

__global__ void k_logsoftmax4(const float* __restrict__ in, float* __restrict__ out, int n) {
    int i = blockIdx.x * blockDim.x + threadIdx.x;
    if (i >= n) return;
    float4 v = ((const float4*)in)[i];
    float m = fmaxf(fmaxf(v.x, v.y), fmaxf(v.z, v.w));
    float s = expf(v.x - m) + expf(v.y - m) + expf(v.z - m) + expf(v.w - m);
    float lse = m + logf(s);
    float4 o = {v.x - lse, v.y - lse, v.z - lse, v.w - lse};
    ((float4*)out)[i] = o;
}

// --------------------------------- host side -------------------------------

static inline int divup(long long a, int b) { return (int)((a + b - 1) / b); }

template <int FIN, int FOUT, bool RELU>
static void run_layer(const float* h_in, const float* W, const float* bias, float* h_out,
                      const int* src32, const int* dst32, const float* norm,
                      float* tx1, float* tx2, float* wpack, int N, int E, hipStream_t stream) {
    const long long nf = (long long)N * FIN;
    const long long sc = (long long)E * (FIN / 4);
    constexpr int NT = (FOUT + 15) / 16;
    // Tx1 = L~ h = -scatter(norm * h[src] -> dst)
    k_zero4<<<divup(nf / 4, 256), 256, 0, stream>>>((float4*)tx1, nf / 4);
    k_scatter<FIN><<<divup(sc, 256), 256, 0, stream>>>(src32, dst32, norm, h_in, tx1, -1.0f, E);
    // Tx2 = 2 L~ Tx1 - Tx0  (init -Tx0, scatter with scale -2)
    k_negcopy4<<<divup(nf / 4, 256), 256, 0, stream>>>((float4*)tx2, (const float4*)h_in, nf / 4);
    k_scatter<FIN><<<divup(sc, 256), 256, 0, stream>>>(src32, dst32, norm, tx1, tx2, -2.0f, E);
    // Pack B fragments, then dense WMMA GEMM
    const int pk = 3 * (FIN / 4) * NT * 32;
    k_pack_w<FIN, FOUT><<<divup(pk, 256), 256, 0, stream>>>(W, (float2*)wpack);
    // accumulators = MR*NT*8 VGPRs; cap at 64 to avoid spills
    constexpr int MR = (NT >= 4) ? 2 : 4;
    const int mtiles = (N + 15) / 16;
    const int groups = divup(mtiles, MR);
    const int wpb = 4;  // waves per block (128 threads)
    k_cheb_gemm<FIN, FOUT, RELU, MR><<<divup(groups, wpb), wpb * 32, 0, stream>>>(
        h_in, tx1, tx2, wpack, bias, h_out, N);
}

extern "C" void kernel_launch(void* const* d_in, const int* in_sizes, int n_in,
                              void* d_out, int out_size, void* d_ws, size_t ws_size,
                              hipStream_t stream) {
    const float* x  = (const float*)d_in[0];
    const int*   ei = (const int*)d_in[1];  // int64 edge_index [2, E]; low words used
    const float* ew = (const float*)d_in[2];
    const float* W1 = (const float*)d_in[3];
    const float* b1 = (const float*)d_in[4];
    const float* W2 = (const float*)d_in[5];
    const float* b2 = (const float*)d_in[6];
    const float* W3 = (const float*)d_in[7];
    const float* b3 = (const float*)d_in[8];
    const float* W4 = (const float*)d_in[9];
    const float* b4 = (const float*)d_in[10];

    const int N = in_sizes[0] / 128;  // 100000
    const int E = in_sizes[2];        // 1600000
    const int* src32 = ei;                          // low words of src int64s
    const int* dst32 = ei + (long long)E * 2;       // low words of dst int64s

    // workspace layout (floats)
    float* ws    = (float*)d_ws;
    float* deg   = ws;  ws += N;                   // then holds dinv (in-place)
    float* norm  = ws;  ws += E;
    float* tx1   = ws;  ws += (long long)N * 128;
    float* tx2   = ws;  ws += (long long)N * 128;
    float* h1    = ws;  ws += (long long)N * 64;
    float* h2    = ws;  ws += (long long)N * 32;
    float* h3    = ws;  ws += (long long)N * 16;
    float* h4    = ws;  ws += (long long)N * 4;
    float* wpack = ws;  ws += 3 * 32 * 4 * 32 * 2;  // max packed size (layer 1)

    // normalization: deg -> dinv -> per-edge norm
    k_zero4<<<divup(N / 4, 256), 256, 0, stream>>>((float4*)deg, N / 4);
    k_deg<<<divup(E, 256), 256, 0, stream>>>(src32, ew, deg, E);
    k_dinv<<<divup(N, 256), 256, 0, stream>>>(deg, N);
    k_edge_norm<<<divup(E, 256), 256, 0, stream>>>(src32, dst32, ew, deg, norm, E);

    // 4 ChebConv layers
    run_layer<128, 64, true >(x,  W1, b1, h1, src32, dst32, norm, tx1, tx2, wpack, N, E, stream);
    run_layer< 64, 32, true >(h1, W2, b2, h2, src32, dst32, norm, tx1, tx2, wpack, N, E, stream);
    run_layer< 32, 16, true >(h2, W3, b3, h3, src32, dst32, norm, tx1, tx2, wpack, N, E, stream);
    run_layer< 16,  4, false>(h3, W4, b4, h4, src32, dst32, norm, tx1, tx2, wpack, N, E, stream);

    // final log_softmax over the 4 class logits
    k_logsoftmax4<<<divup(N, 256), 256, 0, stream>>>(h4, (float*)d_out, N);
}